// Graph_Conv_5231270167040
// MI455X (gfx1250) — compile-verified
//
#include <hip/hip_runtime.h>
#include <math.h>

#define D_IN  128
#define D_OUT 128

typedef float v2f __attribute__((ext_vector_type(2)));
typedef float v8f __attribute__((ext_vector_type(8)));

// ---------------------------------------------------------------------------
// Kernel 1: y = x @ W^T + b using V_WMMA_F32_16X16X4_F32.
// Block = 256 threads = 8 waves; block computes 16 nodes x 128 outputs,
// wave w owns output columns [16w, 16w+16). Two accumulator chains hide the
// WMMA->WMMA RAW hazard. Store tail guard is block-uniform (branch-free fast
// path; N % 16 == 0 in practice).
// ---------------------------------------------------------------------------
__global__ void gemm_bias_wmma_kernel(const float* __restrict__ x,
                                      const float* __restrict__ W,
                                      const float* __restrict__ b,
                                      float* __restrict__ y,
                                      int n_nodes) {
    const int lane = threadIdx.x & 31;
    const int wave = threadIdx.x >> 5;          // 0..7
    const int lm   = lane & 15;                 // M index (A) / N index (B,C,D)
    const int lg   = lane >> 4;                 // lane group: K pair select
    const int row0 = blockIdx.x * 16;           // node tile base
    const int col0 = wave * 16;                 // output-column tile base

    // Clamp loads so EXEC stays all-ones through the WMMA (ISA requirement).
    int arow = row0 + lm;
    if (arow >= n_nodes) arow = n_nodes - 1;

    const float* __restrict__ xptr = x + (size_t)arow * D_IN + 2 * lg;
    const float* __restrict__ wptr = W + (size_t)(col0 + lm) * D_IN + 2 * lg;

    const float bias = b[col0 + lm];
    v8f acc0, acc1;
#pragma unroll
    for (int i = 0; i < 8; ++i) { acc0[i] = bias; acc1[i] = 0.0f; }

#pragma unroll 4
    for (int k = 0; k < D_IN; k += 8) {
        v2f a0 = *(const v2f*)(xptr + k);
        v2f b0 = *(const v2f*)(wptr + k);
        v2f a1 = *(const v2f*)(xptr + k + 4);
        v2f b1 = *(const v2f*)(wptr + k + 4);
        acc0 = __builtin_amdgcn_wmma_f32_16x16x4_f32(false, a0, false, b0,
                                                     (short)0, acc0, false, false);
        acc1 = __builtin_amdgcn_wmma_f32_16x16x4_f32(false, a1, false, b1,
                                                     (short)0, acc1, false, false);
    }

    float* __restrict__ yout = y + (size_t)row0 * D_OUT + col0 + lm;
    if (row0 + 16 <= n_nodes) {           // uniform fast path: no exec juggling
#pragma unroll
        for (int i = 0; i < 8; ++i)
            yout[(size_t)(i + 8 * lg) * D_OUT] = acc0[i] + acc1[i];
    } else {
#pragma unroll
        for (int i = 0; i < 8; ++i) {
            const int m = i + 8 * lg;
            if (row0 + m < n_nodes)
                yout[(size_t)m * D_OUT] = acc0[i] + acc1[i];
        }
    }
}

// ---------------------------------------------------------------------------
// CSR-style binning of edges by row (avoids 400M+ f32 scatter atomics).
// ---------------------------------------------------------------------------
__global__ void zero_i32_kernel(int* __restrict__ p, int n) {
    const int i = blockIdx.x * blockDim.x + threadIdx.x;
    if (i < n) p[i] = 0;
}

__global__ void count_rows_kernel(const int* __restrict__ rows,
                                  int* __restrict__ counts, int n_edges) {
    const int e = blockIdx.x * blockDim.x + threadIdx.x;
    if (e < n_edges) atomicAdd(&counts[rows[e]], 1);
}

// Single-block exclusive scan: 1024 threads = 32 waves (wave32); shuffle scan
// per wave, wave 0 scans the 32 wave sums, LDS carry across 1024-elem chunks.
__global__ void scan_kernel(const int* __restrict__ counts,
                            int* __restrict__ offsets, int n) {
    __shared__ int wsum[32];
    __shared__ int carry_s;
    const int tid  = threadIdx.x;
    const int lane = tid & 31;
    const int wid  = tid >> 5;
    if (tid == 0) carry_s = 0;
    __syncthreads();
    for (int base = 0; base < n; base += 1024) {
        const int idx = base + tid;
        const int v = (idx < n) ? counts[idx] : 0;
        int inc = v;
#pragma unroll
        for (int d = 1; d < 32; d <<= 1) {
            int t = __shfl_up(inc, d, 32);
            if (lane >= d) inc += t;
        }
        if (lane == 31) wsum[wid] = inc;
        __syncthreads();
        if (wid == 0) {
            const int s = wsum[lane];
            int si = s;
#pragma unroll
            for (int d = 1; d < 32; d <<= 1) {
                int t = __shfl_up(si, d, 32);
                if (lane >= d) si += t;
            }
            wsum[lane] = si - s;   // exclusive wave offset
        }
        __syncthreads();
        const int excl = carry_s + wsum[wid] + (inc - v);
        if (idx < n) offsets[idx] = excl;
        __syncthreads();
        if (tid == 1023) carry_s = excl + v;   // carry + chunk total
        __syncthreads();
    }
    if (tid == 0) offsets[n] = carry_s;
}

__global__ void fill_bins_kernel(const int* __restrict__ rows,
                                 const int* __restrict__ cols,
                                 const float* __restrict__ vals,
                                 const int* __restrict__ offsets,
                                 int* __restrict__ cursor,
                                 int* __restrict__ ecol,
                                 float* __restrict__ evalv,
                                 int n_edges) {
    const int e = blockIdx.x * blockDim.x + threadIdx.x;
    if (e < n_edges) {
        const int r = rows[e];
        const int p = offsets[r] + atomicAdd(&cursor[r], 1);
        ecol[p]  = cols[e];
        evalv[p] = vals[e];
    }
}

// ---------------------------------------------------------------------------
// Kernel: per-row gather-sum + fused ELU. One WAVE per row: lane owns 4
// output columns (float4 -> global_load_b128, one vmem instr per edge).
// (col,val) for 32 edges are loaded coalesced once, then broadcast lane-by-
// lane with wave32 __shfl — no per-thread broadcast loads in the hot loop.
// y (51.2 MB) lives in the 192 MB L2, so all gathers are on-chip.
// ---------------------------------------------------------------------------
__global__ void spmm_elu_kernel(const float* __restrict__ y,
                                const int* __restrict__ offsets,
                                const int* __restrict__ ecol,
                                const float* __restrict__ evalv,
                                float* __restrict__ out, int n_rows) {
    const int lane = threadIdx.x & 31;
    const int wid  = threadIdx.x >> 5;          // wave in block (8 waves)
    const int r    = blockIdx.x * 8 + wid;      // one row per wave
    if (r >= n_rows) return;

    const int e_beg = offsets[r];
    const int e_end = offsets[r + 1];
    const float* __restrict__ ycol = y + 4 * lane;

    float a0 = 0.f, a1 = 0.f, a2 = 0.f, a3 = 0.f;

    for (int eb = e_beg; eb < e_end; eb += 32) {
        const int me = eb + lane;
        int   cl = 0;
        float vl = 0.f;
        if (me < e_end) { cl = ecol[me]; vl = evalv[me]; }
        const int cnt = (e_end - eb < 32) ? (e_end - eb) : 32;

        int t = 0;
        for (; t + 4 <= cnt; t += 4) {
            const int   c0 = __shfl(cl, t + 0, 32), c1 = __shfl(cl, t + 1, 32);
            const int   c2 = __shfl(cl, t + 2, 32), c3 = __shfl(cl, t + 3, 32);
            const float v0 = __shfl(vl, t + 0, 32), v1 = __shfl(vl, t + 1, 32);
            const float v2 = __shfl(vl, t + 2, 32), v3 = __shfl(vl, t + 3, 32);
            const float4 y0 = *(const float4*)(ycol + (size_t)c0 * D_OUT);
            const float4 y1 = *(const float4*)(ycol + (size_t)c1 * D_OUT);
            const float4 y2 = *(const float4*)(ycol + (size_t)c2 * D_OUT);
            const float4 y3 = *(const float4*)(ycol + (size_t)c3 * D_OUT);
            a0 = fmaf(v0, y0.x, a0); a1 = fmaf(v0, y0.y, a1);
            a2 = fmaf(v0, y0.z, a2); a3 = fmaf(v0, y0.w, a3);
            a0 = fmaf(v1, y1.x, a0); a1 = fmaf(v1, y1.y, a1);
            a2 = fmaf(v1, y1.z, a2); a3 = fmaf(v1, y1.w, a3);
            a0 = fmaf(v2, y2.x, a0); a1 = fmaf(v2, y2.y, a1);
            a2 = fmaf(v2, y2.z, a2); a3 = fmaf(v2, y2.w, a3);
            a0 = fmaf(v3, y3.x, a0); a1 = fmaf(v3, y3.y, a1);
            a2 = fmaf(v3, y3.z, a2); a3 = fmaf(v3, y3.w, a3);
        }
        for (; t < cnt; ++t) {
            const int   c = __shfl(cl, t, 32);
            const float v = __shfl(vl, t, 32);
            const float4 yv = *(const float4*)(ycol + (size_t)c * D_OUT);
            a0 = fmaf(v, yv.x, a0); a1 = fmaf(v, yv.y, a1);
            a2 = fmaf(v, yv.z, a2); a3 = fmaf(v, yv.w, a3);
        }
    }

    // ELU (alpha = 1)
    float4 o;
    o.x = (a0 > 0.f) ? a0 : expm1f(a0);
    o.y = (a1 > 0.f) ? a1 : expm1f(a1);
    o.z = (a2 > 0.f) ? a2 : expm1f(a2);
    o.w = (a3 > 0.f) ? a3 : expm1f(a3);
    *(float4*)(out + (size_t)r * D_OUT + 4 * lane) = o;
}

// ---------------------------------------------------------------------------
extern "C" void kernel_launch(void* const* d_in, const int* in_sizes, int n_in,
                              void* d_out, int out_size, void* d_ws, size_t ws_size,
                              hipStream_t stream) {
    const float* x    = (const float*)d_in[0];
    const float* W    = (const float*)d_in[1];
    const float* b    = (const float*)d_in[2];
    const int*   rows = (const int*)  d_in[3];
    const int*   cols = (const int*)  d_in[4];
    const float* vals = (const float*)d_in[5];

    const int N = in_sizes[0] / D_IN;   // 100000 nodes
    const int E = in_sizes[3];          // 3200000 edges

    // Workspace carve-out (256B aligned): y | counts+cursor | offsets | ecol | eval
    char*  base = (char*)d_ws;
    size_t off  = 0;
    auto carve = [&](size_t bytes) -> void* {
        void* p = base + off;
        off = (off + bytes + 255) & ~(size_t)255;
        return p;
    };
    float* y      = (float*)carve((size_t)N * D_OUT * sizeof(float));
    int*   counts = (int*)  carve((size_t)2 * N * sizeof(int));
    int*   cursor = counts + N;
    int*   offs   = (int*)  carve((size_t)(N + 1) * sizeof(int));
    int*   ecol   = (int*)  carve((size_t)E * sizeof(int));
    float* evalv  = (float*)carve((size_t)E * sizeof(float));
    (void)ws_size; (void)n_in; (void)out_size;

    // 1. Dense GEMM + bias via f32 WMMA.
    gemm_bias_wmma_kernel<<<(N + 15) / 16, 256, 0, stream>>>(x, W, b, y, N);

    // 2. Bin edges by destination row (CSR build).
    zero_i32_kernel<<<(2 * N + 255) / 256, 256, 0, stream>>>(counts, 2 * N);
    count_rows_kernel<<<(E + 255) / 256, 256, 0, stream>>>(rows, counts, E);
    scan_kernel<<<1, 1024, 0, stream>>>(counts, offs, N);
    fill_bins_kernel<<<(E + 255) / 256, 256, 0, stream>>>(rows, cols, vals,
                                                          offs, cursor, ecol, evalv, E);

    // 3. Row-parallel (wave-per-row) gather-sum + ELU.
    spmm_elu_kernel<<<(N + 7) / 8, 256, 0, stream>>>(y, offs, ecol, evalv,
                                                     (float*)d_out, N);
}